// ScaledDotProductAttention_6193342841468
// MI455X (gfx1250) — compile-verified
//
#include <hip/hip_runtime.h>
#include <hip/hip_bf16.h>

typedef __attribute__((ext_vector_type(2))) float v2f;
typedef __attribute__((ext_vector_type(8))) float v8f;

#define S_DIM 2048
#define DK 128
#define QT 16
#define NWAVE 8
#define SCORE_STRIDE (S_DIM + 2)   // 2050: lane row-stride mod 64 banks = 2 -> conflict-free frags
#define QK_STRIDE (DK + 2)         // 130:  same trick for Q/K tiles
#define NEG_INF (-1e9f)
#define SCALE 0.08838834764831845f // 1/sqrt(128)

// LDS (floats): scores[16][2050] | Q[16][130] | Kstage[8][16][130]
#define SMEM_FLOATS (QT * SCORE_STRIDE + QT * QK_STRIDE + NWAVE * QT * QK_STRIDE)

__launch_bounds__(256)
__global__ void sdpa_wmma_f32_kernel(const float* __restrict__ Q,
                                     const float* __restrict__ K,
                                     const float* __restrict__ V,
                                     const unsigned char* __restrict__ mask,
                                     const float* __restrict__ lps,
                                     const float* __restrict__ conv_w,
                                     const float* __restrict__ conv_b,
                                     float* __restrict__ ctx_out,
                                     float* __restrict__ attn_out) {
  extern __shared__ float smem[];
  float* sScores = smem;                                // 16 x 2050
  float* sQ = sScores + QT * SCORE_STRIDE;              // 16 x 130
  float* sK = sQ + QT * QK_STRIDE;                      // 8 waves x 16 x 130

  const int tid  = threadIdx.x;
  const int lane = tid & 31;
  const int wave = tid >> 5;

  const int qTile = blockIdx.x & ((S_DIM / QT) - 1);    // 0..127
  const int bh    = blockIdx.x >> 7;                    // 0..31  (B*H)
  const int qbase = qTile * QT;

  const float w = conv_w[0];
  const float bc = conv_b[0];

  const float* Qbh = Q + ((size_t)bh * S_DIM + qbase) * DK;
  const float* Kbh = K + (size_t)bh * S_DIM * DK;
  const float* Vbh = V + (size_t)bh * S_DIM * DK;
  const unsigned char* Mbh = mask + ((size_t)bh * S_DIM + qbase) * S_DIM;
  float* ctx   = ctx_out  + ((size_t)bh * S_DIM + qbase) * DK;
  float* attnO = attn_out + ((size_t)bh * S_DIM + qbase) * S_DIM;

  // ---- Stage Q tile (16x128) into LDS, coalesced float4 ----
  for (int i = tid; i < QT * DK / 4; i += 256) {
    int r = i >> 5;            // DK/4 = 32 float4 per row
    int c4 = i & 31;
    float4 v = ((const float4*)(Qbh + (size_t)r * DK))[c4];
    float* dst = sQ + r * QK_STRIDE + c4 * 4;
    dst[0] = v.x; dst[1] = v.y; dst[2] = v.z; dst[3] = v.w;
  }
  __syncthreads();

  const int mrow = lane & 15;          // M (A-frag) / N (B-frag) index
  const int halfsel = (lane >> 4) * 2; // lanes 16-31 hold K+2/K+3
  const int rowoff = (lane >> 4) * 8;  // C/D: high lanes hold M+8

  // ---- Phase 1: scores = QK^T * scale + (w*lps + b), mask -> LDS ----
  float* sKw = sK + wave * QT * QK_STRIDE;
  for (int kt = wave; kt < S_DIM / 16; kt += NWAVE) {
    const float* Kt = Kbh + (size_t)kt * 16 * DK;
    // wave-private stage of K tile (16x128), coalesced float4
    for (int i = lane; i < QT * DK / 4; i += 32) {
      int r = i >> 5;
      int c4 = i & 31;
      float4 v = ((const float4*)(Kt + (size_t)r * DK))[c4];
      float* dst = sKw + r * QK_STRIDE + c4 * 4;
      dst[0] = v.x; dst[1] = v.y; dst[2] = v.z; dst[3] = v.w;
    }

    v8f acc = {0.f, 0.f, 0.f, 0.f, 0.f, 0.f, 0.f, 0.f};
#pragma unroll 8
    for (int d0 = 0; d0 < DK; d0 += 4) {
      const float* ap = sQ  + mrow * QK_STRIDE + d0 + halfsel;
      const float* bp = sKw + mrow * QK_STRIDE + d0 + halfsel;
      v2f a, b;
      a.x = ap[0]; a.y = ap[1];
      b.x = bp[0]; b.y = bp[1];
      acc = __builtin_amdgcn_wmma_f32_16x16x4_f32(false, a, false, b,
                                                  (short)0, acc, false, false);
    }

    const int colbase = kt * 16;
    const int ncol = lane & 15;
#pragma unroll
    for (int j = 0; j < 8; ++j) {
      int m = j + rowoff;
      float node = w * lps[(size_t)(qbase + m) * S_DIM + colbase + ncol] + bc;
      float sc = acc[j] * SCALE + node;
      if (Mbh[(size_t)m * S_DIM + colbase + ncol]) sc = NEG_INF;
      sScores[m * SCORE_STRIDE + colbase + ncol] = sc;
    }
  }
  __syncthreads();

  // ---- Phase 2: softmax over rows (one wave handles 2 rows) ----
  for (int rr = 0; rr < 2; ++rr) {
    const int m = wave * 2 + rr;
    float* row = sScores + m * SCORE_STRIDE;
    float mx = -3.4e38f;
    for (int c = lane; c < S_DIM; c += 32) mx = fmaxf(mx, row[c]);
#pragma unroll
    for (int off = 16; off > 0; off >>= 1) mx = fmaxf(mx, __shfl_xor(mx, off, 32));
    float sum = 0.f;
    for (int c = lane; c < S_DIM; c += 32) {
      float e = __expf(row[c] - mx);
      row[c] = e;
      sum += e;
    }
#pragma unroll
    for (int off = 16; off > 0; off >>= 1) sum += __shfl_xor(sum, off, 32);
    const float inv = 1.0f / sum;
    float* arow = attnO + (size_t)m * S_DIM;
    for (int c = lane; c < S_DIM; c += 32) {
      float a = row[c] * inv;
      row[c] = a;          // keep normalized probs for the PV matmul
      arow[c] = a;         // stream attn output (coalesced)
    }
  }
  __syncthreads();

  // ---- Phase 3: context = attn @ V, each wave owns 16 output columns ----
  {
    const int nbase = wave * 16;
    const int ncol = lane & 15;
    v8f acc = {0.f, 0.f, 0.f, 0.f, 0.f, 0.f, 0.f, 0.f};
#pragma unroll 4
    for (int k0 = 0; k0 < S_DIM; k0 += 4) {
      const float* ap = sScores + mrow * SCORE_STRIDE + k0 + halfsel;
      const float* vp = Vbh + (size_t)(k0 + halfsel) * DK + nbase + ncol;
      v2f a, b;
      a.x = ap[0]; a.y = ap[1];
      b.x = vp[0]; b.y = vp[DK];
      acc = __builtin_amdgcn_wmma_f32_16x16x4_f32(false, a, false, b,
                                                  (short)0, acc, false, false);
    }
#pragma unroll
    for (int j = 0; j < 8; ++j) {
      int m = j + rowoff;
      ctx[(size_t)m * DK + nbase + ncol] = acc[j];
    }
  }
}

extern "C" void kernel_launch(void* const* d_in, const int* in_sizes, int n_in,
                              void* d_out, int out_size, void* d_ws, size_t ws_size,
                              hipStream_t stream) {
  const float* Q = (const float*)d_in[0];
  const float* K = (const float*)d_in[1];
  const float* V = (const float*)d_in[2];
  const unsigned char* mask = (const unsigned char*)d_in[3];
  const float* lps = (const float*)d_in[4];
  const float* cw = (const float*)d_in[5];
  const float* cb = (const float*)d_in[6];

  // B*H derived from Q's flat size: BH = |Q| / (S*Dk)
  const int BH = in_sizes[0] / (S_DIM * DK);
  const int blocks = BH * (S_DIM / QT);     // 32 * 128 = 4096

  float* ctx_out = (float*)d_out;                                    // [B,H,S,Dk]
  float* attn_out = ctx_out + (size_t)BH * S_DIM * DK;               // [B,H,S,S]

  const size_t smem_bytes = (size_t)SMEM_FLOATS * sizeof(float);     // ~201 KB of 320 KB WGP LDS

  sdpa_wmma_f32_kernel<<<dim3(blocks), dim3(256), smem_bytes, stream>>>(
      Q, K, V, mask, lps, cw, cb, ctx_out, attn_out);
}